// MASK_80341658239303
// MI455X (gfx1250) — compile-verified
//
#include <hip/hip_runtime.h>
#include <stdint.h>

// Pure bandwidth-bound masked-unfold gather.
// x: (8,64,256,256) f32  ->  out: (8,64,264,248) f32
// CDNA5 path: global_load_async_to_lds_b128 staging + s_wait_asynccnt,
// coalesced float4 stores.

#define HAVE_ASYNC_LDS __has_builtin(__builtin_amdgcn_global_load_async_to_lds_b128)
#define HAVE_WAIT_ASYNC __has_builtin(__builtin_amdgcn_s_wait_asynccnt)

// Builtin signature (from hipcc diagnostic): param 1 is
// 'int __attribute__((vector_size(16))) __device__ *' i.e. global int4*.
typedef int vint4 __attribute__((vector_size(4 * sizeof(int))));
typedef __attribute__((address_space(1))) vint4 gvint4;  // global
typedef __attribute__((address_space(3))) vint4 lvint4;  // LDS

__global__ __launch_bounds__(256) void MASK_80341658239303_kernel(
    const float* __restrict__ x, float* __restrict__ out) {
  // 8 waves/block, each wave stages one 256-float source row (1 KB slice).
  __shared__ __align__(16) float lds[8 * 256];

  const int hh   = blockIdx.x;   // 0..32  (output row group)
  const int c    = blockIdx.y;   // 0..63
  const int b    = blockIdx.z;   // 0..7
  const int tid  = threadIdx.x;
  const int wave = tid >> 5;     // = jr, output row within group
  const int lane = tid & 31;

  const int jr   = wave;
  const int offh = 2 * jr + (jr >= 4 ? 1 : 0);       // masked H-window offset
  const int hp   = hh * 8 + offh;                    // padded-H index
  const bool zero_row = (hp < 8);                    // top zero padding
  const int hsrc = (hp < 264) ? (hp - 8) : (hp - 16); // reflect bottom rows

  const int bc = b * 64 + c;
  float* lrow = &lds[wave * 256];

  if (!zero_row) {
    const float* src = x + ((long)bc * 256 + hsrc) * 256;  // 1 KB, 16B-aligned
#if HAVE_ASYNC_LDS
    // Two b128 async bursts per lane: 32 lanes * 16B * 2 = 1024B (full row).
    __builtin_amdgcn_global_load_async_to_lds_b128(
        (gvint4*)(src + lane * 4),
        (lvint4*)(lrow + lane * 4), 0, 0);
    __builtin_amdgcn_global_load_async_to_lds_b128(
        (gvint4*)(src + lane * 4 + 128),
        (lvint4*)(lrow + lane * 4 + 128), 0, 0);
#if HAVE_WAIT_ASYNC
    __builtin_amdgcn_s_wait_asynccnt(0);
#else
    asm volatile("s_wait_asynccnt 0" ::: "memory");
#endif
#else
    // Fallback: VGPR staging (still fully coalesced b128 loads).
    const float4* s4 = (const float4*)src;
    float4*       l4 = (float4*)lrow;
    l4[lane]      = s4[lane];
    l4[lane + 32] = s4[lane + 32];
#endif
  }
  // No block barrier needed: each wave reads only the LDS slice it loaded.

  const int r = hh * 8 + jr;                          // output row 0..263
  float* orow = out + ((long)bc * 264 + r) * 248;
  for (int t = lane; t < 62; t += 32) {               // 62 float4 per row
    const int cc0 = t * 4;
    float4 v;
    if (zero_row) {
      v = make_float4(0.f, 0.f, 0.f, 0.f);
    } else {
      float e[4];
#pragma unroll
      for (int k = 0; k < 4; ++k) {
        const int cc = cc0 + k;
        const int ic = cc & 7;
        const int wp = (cc & ~7) + 2 * ic + (ic >= 4 ? 1 : 0);
        const int ws = (wp < 248) ? wp : (wp - 8);    // reflect right cols
        e[k] = lrow[ws];
      }
      v = make_float4(e[0], e[1], e[2], e[3]);
    }
    *(float4*)(orow + cc0) = v;                       // coalesced b128 store
  }
}

extern "C" void kernel_launch(void* const* d_in, const int* in_sizes, int n_in,
                              void* d_out, int out_size, void* d_ws, size_t ws_size,
                              hipStream_t stream) {
  const float* x = (const float*)d_in[0];
  float* out = (float*)d_out;
  // grid: hh groups (33) x channels (64) x batch (8); 256 threads = 8 waves
  dim3 grid(33, 64, 8);
  dim3 block(256);
  MASK_80341658239303_kernel<<<grid, block, 0, stream>>>(x, out);
}